// VoteGCL_Encoder_89008902242435
// MI455X (gfx1250) — compile-verified
//
#include <hip/hip_runtime.h>
#include <cstdint>

#define USER_NUM 200000
#define ITEM_NUM 100000
#define N_NODES (USER_NUM + ITEM_NUM)
#define EMB 64
#define N_LAYERS 3

static constexpr int EPB = 256;  // edges staged per block

// ---- ego0 = concat(user, item); acc = ego0; ego1 = 0 (first scatter target) --
__global__ __launch_bounds__(256) void init_concat_zero(const float4* __restrict__ ue,
                                                        const float4* __restrict__ ie,
                                                        float4* __restrict__ ego,
                                                        float4* __restrict__ acc,
                                                        float4* __restrict__ zbuf,
                                                        int n_user4, int n_total4) {
  int i = blockIdx.x * blockDim.x + threadIdx.x;
  if (i >= n_total4) return;
  float4 v = (i < n_user4) ? ue[i] : ie[i - n_user4];
  ego[i]  = v;
  acc[i]  = v;
  zbuf[i] = make_float4(0.f, 0.f, 0.f, 0.f);
}

// acc = (acc + x) * s ; optionally zero z (next layer's scatter target, which
// was only *read* by the SpMM that produced x, so this is stream-order safe).
__global__ __launch_bounds__(256) void add_scale_zero4(float4* __restrict__ acc,
                                                       const float4* __restrict__ x,
                                                       float4* __restrict__ z,
                                                       float s, int n4) {
  int i = blockIdx.x * blockDim.x + threadIdx.x;
  if (i >= n4) return;
  float4 a = acc[i];
  float4 b = x[i];
  a.x = (a.x + b.x) * s;
  a.y = (a.y + b.y) * s;
  a.z = (a.z + b.z) * s;
  a.w = (a.w + b.w) * s;
  acc[i] = a;
  if (z) z[i] = make_float4(0.f, 0.f, 0.f, 0.f);   // uniform branch
}

// ---- COO SpMM scatter: y[rows[e]] += vals[e] * x[cols[e]] -------------------
// Block stages EPB edges (rows/cols/vals) into LDS via CDNA5 async-to-LDS DMA
// (ASYNCcnt path), then 16 lanes cooperate per edge: one float4 (16 B) of the
// 256 B embedding row each. Gathers hit the 192 MB L2 (x table = 76.8 MB);
// scatter uses no-return global_atomic_add_f32 resolved at L2.
__global__ __launch_bounds__(256) void spmm_scatter(const float* __restrict__ vals,
                                                    const int* __restrict__ rows,
                                                    const int* __restrict__ cols,
                                                    const float* __restrict__ x,
                                                    float* __restrict__ y,
                                                    int nnz) {
  __shared__ int   s_rows[EPB];
  __shared__ int   s_cols[EPB];
  __shared__ float s_vals[EPB];

  const int tid  = threadIdx.x;
  const int base = blockIdx.x * EPB;
  int n = nnz - base;
  if (n <= 0) return;
  if (n > EPB) n = EPB;

  if (n == EPB) {
    // Async DMA the three contiguous edge streams into LDS.
    uint32_t l_r  = (uint32_t)(size_t)&s_rows[tid];
    uint32_t l_c  = (uint32_t)(size_t)&s_cols[tid];
    uint32_t l_v  = (uint32_t)(size_t)&s_vals[tid];
    uint32_t goff = (uint32_t)(base + tid) * 4u;
    asm volatile(
        "global_load_async_to_lds_b32 %0, %3, %4\n\t"
        "global_load_async_to_lds_b32 %1, %3, %5\n\t"
        "global_load_async_to_lds_b32 %2, %3, %6\n\t"
        "s_wait_asynccnt 0"
        :
        : "v"(l_r), "v"(l_c), "v"(l_v), "v"(goff),
          "s"(rows), "s"(cols), "s"(vals)
        : "memory");
  } else {
    if (tid < n) {
      s_rows[tid] = rows[base + tid];
      s_cols[tid] = cols[base + tid];
      s_vals[tid] = vals[base + tid];
    }
  }
  __syncthreads();

  const int lane16 = tid & 15;   // which 16 B slice of the 256 B row
  const int grp    = tid >> 4;   // 16 edge-groups per block
  const float4* x4 = (const float4*)x;

  for (int e = grp; e < n; e += 16) {
    const int   r = s_rows[e];   // LDS broadcast within the 16-lane group
    const int   c = s_cols[e];
    const float v = s_vals[e];
    float4 g = x4[(size_t)c * (EMB / 4) + lane16];
    float* yp = y + (size_t)r * EMB + (lane16 << 2);
    unsafeAtomicAdd(yp + 0, v * g.x);
    unsafeAtomicAdd(yp + 1, v * g.y);
    unsafeAtomicAdd(yp + 2, v * g.z);
    unsafeAtomicAdd(yp + 3, v * g.w);
  }
}

extern "C" void kernel_launch(void* const* d_in, const int* in_sizes, int n_in,
                              void* d_out, int out_size, void* d_ws, size_t ws_size,
                              hipStream_t stream) {
  const float* ue   = (const float*)d_in[0];
  const float* ie   = (const float*)d_in[1];
  const float* vals = (const float*)d_in[2];
  const int*   rows = (const int*)d_in[3];
  const int*   cols = (const int*)d_in[4];
  const int    nnz  = in_sizes[2];

  float* acc  = (float*)d_out;                       // running sum, ends as mean
  float* ego0 = (float*)d_ws;                        // ping
  float* ego1 = ego0 + (size_t)N_NODES * EMB;        // pong

  const int n_total4 = N_NODES * EMB / 4;            // 4.8M float4
  const int n_user4  = USER_NUM * EMB / 4;
  const int eb = 256;
  const int gE = (n_total4 + eb - 1) / eb;
  const int gS = (nnz + EPB - 1) / EPB;

  init_concat_zero<<<gE, eb, 0, stream>>>((const float4*)ue, (const float4*)ie,
                                          (float4*)ego0, (float4*)acc,
                                          (float4*)ego1, n_user4, n_total4);

  float* cur = ego0;
  float* nxt = ego1;
  for (int l = 0; l < N_LAYERS; ++l) {
    spmm_scatter<<<gS, 256, 0, stream>>>(vals, rows, cols, cur, nxt, nnz);
    const bool  last = (l == N_LAYERS - 1);
    const float s    = last ? (1.0f / (N_LAYERS + 1)) : 1.0f;
    // zero `cur` (just read by spmm) so it is clean as the next scatter target
    add_scale_zero4<<<gE, eb, 0, stream>>>((float4*)acc, (const float4*)nxt,
                                           last ? (float4*)nullptr : (float4*)cur,
                                           s, n_total4);
    float* t = cur; cur = nxt; nxt = t;
  }
}